// LSTMTracker_81200651698400
// MI455X (gfx1250) — compile-verified
//
#include <hip/hip_runtime.h>
#include <hip/hip_bf16.h>

typedef __attribute__((ext_vector_type(2))) float v2f;
typedef __attribute__((ext_vector_type(4))) float v4f;
typedef __attribute__((ext_vector_type(8))) float v8f;

#define IN_DIM   131
#define IN_PAD   132
#define H1D      64
#define G1       256   // 4*H1
#define H2D      32
#define G2       128   // 4*H2
#define T_SEQ    256
#define BATCH    2048
#define NCLS     4
#define MROWS    (BATCH * T_SEQ)   // 524288 flattened [b*T+t] rows

__device__ __forceinline__ float sigmoid_(float x) {
    return 1.0f / (1.0f + __expf(-x));
}

// D = A(16x4) * B(4x16) + C.  a = {A[m][k0],A[m][k0+1]}, b = {B[k0][n],B[k0+1][n]}
__device__ __forceinline__ v8f wmma4(v2f a, v2f b, v8f acc) {
    return __builtin_amdgcn_wmma_f32_16x16x4_f32(
        false, a, false, b, (short)0, acc, false, false);
}

// ===========================================================================
// SPLIT PATH
// ===========================================================================

// ---------------------------------------------------------------------------
// Input-projection GEMM: XP[M][NG] = X[M][K_RAW] @ W[NG][K_RAW]^T + b_ih + b_hh
// 256 threads / 8 waves; 32-row M tiles, grid-stride. Weights staged in LDS in
// k-paired layout so each B fragment is one aligned ds_load_b64.
// ---------------------------------------------------------------------------
template <int K_RAW, int K_PAD, int NG, int NT_W>   // NT_W = NG/64 tiles/wave
__global__ void proj_kernel(const float* __restrict__ X,
                            const float* __restrict__ W,
                            const float* __restrict__ b_ih,
                            const float* __restrict__ b_hh,
                            float* __restrict__ XP, int Mtiles) {
    extern __shared__ float smem[];
    float* wp   = smem;                         // [K_PAD/2][2*NG]
    float* xbuf = smem + (K_PAD / 2) * (2 * NG);  // [32][K_PAD]

    const int tid = threadIdx.x, nthr = blockDim.x;   // 256
    for (int idx = tid; idx < (K_PAD / 2) * (2 * NG); idx += nthr) {
        int kp  = idx / (2 * NG);
        int rem = idx - kp * (2 * NG);
        int n = rem >> 1;
        int k = kp * 2 + (rem & 1);
        wp[idx] = (k < K_RAW) ? W[n * K_RAW + k] : 0.0f;
    }
    __syncthreads();

    const int wave = tid >> 5, lane = tid & 31;
    const int l15 = lane & 15, lhalf = lane >> 4;
    const int mt  = wave >> 2;       // 0..1
    const int nt0 = wave & 3;        // n-tile base, stride 4

    float bias[NT_W];
    #pragma unroll
    for (int j = 0; j < NT_W; ++j) {
        int col = (nt0 + 4 * j) * 16 + l15;
        bias[j] = b_ih[col] + b_hh[col];
    }

    for (int m32 = blockIdx.x; m32 < Mtiles; m32 += gridDim.x) {
        for (int idx = tid; idx < 32 * K_PAD; idx += nthr) {
            int r  = idx / K_PAD;
            int kk = idx - r * K_PAD;
            xbuf[idx] = (kk < K_RAW)
                ? X[(size_t)(m32 * 32 + r) * K_RAW + kk] : 0.0f;
        }
        __syncthreads();

        v8f acc[NT_W];
        #pragma unroll
        for (int j = 0; j < NT_W; ++j) {
            #pragma unroll
            for (int r = 0; r < 8; ++r) acc[j][r] = bias[j];
        }
        for (int ks = 0; ks < K_PAD / 4; ++ks) {
            const int kp = ks * 2 + lhalf;   // pair row
            v2f a = *(const v2f*)(xbuf + (mt * 16 + l15) * K_PAD + kp * 2);
            #pragma unroll
            for (int j = 0; j < NT_W; ++j) {
                int col = (nt0 + 4 * j) * 16 + l15;
                v2f b = *(const v2f*)(wp + kp * (2 * NG) + 2 * col);
                acc[j] = wmma4(a, b, acc[j]);
            }
        }
        #pragma unroll
        for (int j = 0; j < NT_W; ++j) {
            int col = (nt0 + 4 * j) * 16 + l15;
            #pragma unroll
            for (int r = 0; r < 8; ++r) {
                int m = mt * 16 + r + 8 * lhalf;
                XP[(size_t)(m32 * 32 + m) * NG + col] = acc[j][r];
            }
        }
        __syncthreads();
    }
}

// ---------------------------------------------------------------------------
// Layer-1 recurrence only: gates_t = XP1[(b*T+t)] + h_{t-1} @ W_hh^T.
// 128 threads / 4 waves, 16 batch rows per block. 84 KB LDS -> 3 WG/WGP.
// ---------------------------------------------------------------------------
__global__ void rec1_kernel(const float* __restrict__ XP,    // [B*T][256]
                            const float* __restrict__ W_hh,  // [256][64]
                            float* __restrict__ h1out) {     // [B][T][64]
    extern __shared__ float smem[];
    float* whp   = smem;                 // [32][512] k-paired
    float* xpbuf = smem + 32 * 512;      // [16][256]
    float* hbuf  = xpbuf + 16 * G1;      // [16][64]

    const int tid = threadIdx.x, nthr = blockDim.x;  // 128
    for (int idx = tid; idx < 32 * 512; idx += nthr) {
        int kp = idx >> 9, rem = idx & 511;
        int n = rem >> 1, k = kp * 2 + (rem & 1);
        whp[idx] = W_hh[n * H1D + k];
    }
    for (int idx = tid; idx < 16 * H1D; idx += nthr) hbuf[idx] = 0.0f;
    __syncthreads();

    const int wave = tid >> 5, lane = tid & 31;      // wave = hidden subtile
    const int l15 = lane & 15, lhalf = lane >> 4;
    const int hid0 = wave * 16;
    const int b0 = blockIdx.x * 16;

    const int coli = 0 * H1D + hid0 + l15;
    const int colf = 1 * H1D + hid0 + l15;
    const int colg = 2 * H1D + hid0 + l15;
    const int colo = 3 * H1D + hid0 + l15;

    v8f cstate;
    #pragma unroll
    for (int r = 0; r < 8; ++r) cstate[r] = 0.0f;

    for (int t = 0; t < T_SEQ; ++t) {
        // Stage gate pre-activations [16][256] (coalesced float4).
        for (int i = tid; i < 16 * (G1 / 4); i += nthr) {
            int r = i >> 6, c4 = i & 63;
            ((v4f*)xpbuf)[i] =
                ((const v4f*)XP)[((size_t)(b0 + r) * T_SEQ + t) * (G1 / 4) + c4];
        }
        if (t + 1 < T_SEQ) {   // prefetch next step's gate rows
            int r = tid >> 3, seg = tid & 7;
            __builtin_prefetch(XP + ((size_t)(b0 + r) * T_SEQ + (t + 1)) * G1 + seg * 32, 0, 1);
        }
        __syncthreads();

        v8f ai, af, ag, ao;
        #pragma unroll
        for (int r = 0; r < 8; ++r) {
            int m = (r + 8 * lhalf) * G1;
            ai[r] = xpbuf[m + coli];
            af[r] = xpbuf[m + colf];
            ag[r] = xpbuf[m + colg];
            ao[r] = xpbuf[m + colo];
        }
        for (int ks = 0; ks < H1D / 4; ++ks) {       // K = 64
            const int kp = ks * 2 + lhalf;
            v2f a = *(const v2f*)(hbuf + l15 * H1D + kp * 2);
            ai = wmma4(a, *(const v2f*)(whp + kp * 512 + 2 * coli), ai);
            af = wmma4(a, *(const v2f*)(whp + kp * 512 + 2 * colf), af);
            ag = wmma4(a, *(const v2f*)(whp + kp * 512 + 2 * colg), ag);
            ao = wmma4(a, *(const v2f*)(whp + kp * 512 + 2 * colo), ao);
        }
        __syncthreads();

        #pragma unroll
        for (int r = 0; r < 8; ++r) {
            float iv = sigmoid_(ai[r]);
            float fv = sigmoid_(af[r]);
            float gv = tanhf(ag[r]);
            float ov = sigmoid_(ao[r]);
            float cv = fv * cstate[r] + iv * gv;
            cstate[r] = cv;
            hbuf[(r + 8 * lhalf) * H1D + hid0 + l15] = ov * tanhf(cv);
        }
        __syncthreads();

        // Coalesced store of h_t (reads hbuf; safe until next elementwise).
        for (int i = tid; i < 16 * (H1D / 4); i += nthr) {
            int r = i >> 4, c4 = i & 15;
            ((v4f*)h1out)[((size_t)(b0 + r) * T_SEQ + t) * (H1D / 4) + c4] =
                ((const v4f*)hbuf)[i];
        }
    }
}

// ---------------------------------------------------------------------------
// Layer-2 recurrence + FC. 64 threads / 2 waves, 16 batch rows per block.
// ---------------------------------------------------------------------------
__global__ void rec2_fc_kernel(const float* __restrict__ XP,    // [B*T][128]
                               const float* __restrict__ W_hh,  // [128][32]
                               const float* __restrict__ W_fc,  // [4][32]
                               const float* __restrict__ b_fc,  // [4]
                               float* __restrict__ out) {       // [B][4]
    __shared__ float whp[16 * 256];     // [16][256] k-paired (K=32)
    __shared__ float xpbuf[16 * G2];    // [16][128]
    __shared__ float hbuf[16 * H2D];    // [16][32]

    const int tid = threadIdx.x, nthr = blockDim.x;  // 64
    for (int idx = tid; idx < 16 * 256; idx += nthr) {
        int kp = idx >> 8, rem = idx & 255;
        int n = rem >> 1, k = kp * 2 + (rem & 1);
        whp[idx] = W_hh[n * H2D + k];
    }
    for (int idx = tid; idx < 16 * H2D; idx += nthr) hbuf[idx] = 0.0f;
    __syncthreads();

    const int wave = tid >> 5, lane = tid & 31;
    const int l15 = lane & 15, lhalf = lane >> 4;
    const int hid0 = wave * 16;
    const int b0 = blockIdx.x * 16;

    const int coli = 0 * H2D + hid0 + l15;
    const int colf = 1 * H2D + hid0 + l15;
    const int colg = 2 * H2D + hid0 + l15;
    const int colo = 3 * H2D + hid0 + l15;

    v8f cstate;
    #pragma unroll
    for (int r = 0; r < 8; ++r) cstate[r] = 0.0f;

    for (int t = 0; t < T_SEQ; ++t) {
        for (int i = tid; i < 16 * (G2 / 4); i += nthr) {
            int r = i >> 5, c4 = i & 31;
            ((v4f*)xpbuf)[i] =
                ((const v4f*)XP)[((size_t)(b0 + r) * T_SEQ + t) * (G2 / 4) + c4];
        }
        if (t + 1 < T_SEQ) {
            int r = tid >> 2, seg = tid & 3;
            __builtin_prefetch(XP + ((size_t)(b0 + r) * T_SEQ + (t + 1)) * G2 + seg * 32, 0, 1);
        }
        __syncthreads();

        v8f ai, af, ag, ao;
        #pragma unroll
        for (int r = 0; r < 8; ++r) {
            int m = (r + 8 * lhalf) * G2;
            ai[r] = xpbuf[m + coli];
            af[r] = xpbuf[m + colf];
            ag[r] = xpbuf[m + colg];
            ao[r] = xpbuf[m + colo];
        }
        for (int ks = 0; ks < H2D / 4; ++ks) {       // K = 32
            const int kp = ks * 2 + lhalf;
            v2f a = *(const v2f*)(hbuf + l15 * H2D + kp * 2);
            ai = wmma4(a, *(const v2f*)(whp + kp * 256 + 2 * coli), ai);
            af = wmma4(a, *(const v2f*)(whp + kp * 256 + 2 * colf), af);
            ag = wmma4(a, *(const v2f*)(whp + kp * 256 + 2 * colg), ag);
            ao = wmma4(a, *(const v2f*)(whp + kp * 256 + 2 * colo), ao);
        }
        __syncthreads();

        #pragma unroll
        for (int r = 0; r < 8; ++r) {
            float iv = sigmoid_(ai[r]);
            float fv = sigmoid_(af[r]);
            float gv = tanhf(ag[r]);
            float ov = sigmoid_(ao[r]);
            float cv = fv * cstate[r] + iv * gv;
            cstate[r] = cv;
            hbuf[(r + 8 * lhalf) * H2D + hid0 + l15] = ov * tanhf(cv);
        }
        __syncthreads();
    }

    const int m = tid >> 2, cls = tid & 3;
    float s = b_fc[cls];
    #pragma unroll
    for (int j = 0; j < H2D; ++j) s += hbuf[m * H2D + j] * W_fc[cls * H2D + j];
    out[(size_t)(b0 + m) * NCLS + cls] = s;
}

// ===========================================================================
// FUSED FALLBACK PATH (small workspace): proj + recurrence in one kernel,
// upgraded to the k-paired weight layout.
// ===========================================================================
__global__ void fused_layer1_kernel(const float* __restrict__ x,
                                    const float* __restrict__ W_ih,
                                    const float* __restrict__ W_hh,
                                    const float* __restrict__ b_ih,
                                    const float* __restrict__ b_hh,
                                    float* __restrict__ h1out) {
    extern __shared__ float smem[];
    float* wip  = smem;                    // [66][512]  k-paired, K padded 132
    float* whp  = wip + 66 * 512;          // [32][512]
    float* xbuf = whp + 32 * 512;          // [16][132]
    float* hbuf = xbuf + 16 * IN_PAD;      // [16][64]

    const int tid = threadIdx.x, nthr = blockDim.x;   // 128
    for (int idx = tid; idx < 66 * 512; idx += nthr) {
        int kp = idx >> 9, rem = idx & 511;
        int n = rem >> 1, k = kp * 2 + (rem & 1);
        wip[idx] = (k < IN_DIM) ? W_ih[n * IN_DIM + k] : 0.0f;
    }
    for (int idx = tid; idx < 32 * 512; idx += nthr) {
        int kp = idx >> 9, rem = idx & 511;
        int n = rem >> 1, k = kp * 2 + (rem & 1);
        whp[idx] = W_hh[n * H1D + k];
    }
    for (int idx = tid; idx < 16 * H1D; idx += nthr) hbuf[idx] = 0.0f;
    __syncthreads();

    const int wave = tid >> 5, lane = tid & 31;
    const int l15 = lane & 15, lhalf = lane >> 4;
    const int hid0 = wave * 16;
    const int b0 = blockIdx.x * 16;

    const int coli = 0 * H1D + hid0 + l15;
    const int colf = 1 * H1D + hid0 + l15;
    const int colg = 2 * H1D + hid0 + l15;
    const int colo = 3 * H1D + hid0 + l15;

    const float bias_i = b_ih[coli] + b_hh[coli];
    const float bias_f = b_ih[colf] + b_hh[colf];
    const float bias_g = b_ih[colg] + b_hh[colg];
    const float bias_o = b_ih[colo] + b_hh[colo];

    v8f cstate;
    #pragma unroll
    for (int r = 0; r < 8; ++r) cstate[r] = 0.0f;

    for (int t = 0; t < T_SEQ; ++t) {
        for (int idx = tid; idx < 16 * IN_PAD; idx += nthr) {
            int r = idx / IN_PAD, kk = idx - r * IN_PAD;
            xbuf[idx] = (kk < IN_DIM)
                ? x[((size_t)(b0 + r) * T_SEQ + t) * IN_DIM + kk] : 0.0f;
        }
        __syncthreads();

        v8f ai, af, ag, ao;
        #pragma unroll
        for (int r = 0; r < 8; ++r) { ai[r] = bias_i; af[r] = bias_f; ag[r] = bias_g; ao[r] = bias_o; }

        for (int ks = 0; ks < IN_PAD / 4; ++ks) {
            const int kp = ks * 2 + lhalf;
            v2f a = *(const v2f*)(xbuf + l15 * IN_PAD + kp * 2);
            ai = wmma4(a, *(const v2f*)(wip + kp * 512 + 2 * coli), ai);
            af = wmma4(a, *(const v2f*)(wip + kp * 512 + 2 * colf), af);
            ag = wmma4(a, *(const v2f*)(wip + kp * 512 + 2 * colg), ag);
            ao = wmma4(a, *(const v2f*)(wip + kp * 512 + 2 * colo), ao);
        }
        for (int ks = 0; ks < H1D / 4; ++ks) {
            const int kp = ks * 2 + lhalf;
            v2f a = *(const v2f*)(hbuf + l15 * H1D + kp * 2);
            ai = wmma4(a, *(const v2f*)(whp + kp * 512 + 2 * coli), ai);
            af = wmma4(a, *(const v2f*)(whp + kp * 512 + 2 * colf), af);
            ag = wmma4(a, *(const v2f*)(whp + kp * 512 + 2 * colg), ag);
            ao = wmma4(a, *(const v2f*)(whp + kp * 512 + 2 * colo), ao);
        }
        __syncthreads();

        #pragma unroll
        for (int r = 0; r < 8; ++r) {
            float iv = sigmoid_(ai[r]);
            float fv = sigmoid_(af[r]);
            float gv = tanhf(ag[r]);
            float ov = sigmoid_(ao[r]);
            float cv = fv * cstate[r] + iv * gv;
            cstate[r] = cv;
            hbuf[(r + 8 * lhalf) * H1D + hid0 + l15] = ov * tanhf(cv);
        }
        __syncthreads();

        for (int i = tid; i < 16 * (H1D / 4); i += nthr) {
            int r = i >> 4, c4 = i & 15;
            ((v4f*)h1out)[((size_t)(b0 + r) * T_SEQ + t) * (H1D / 4) + c4] =
                ((const v4f*)hbuf)[i];
        }
    }
}

__global__ void fused_layer2_fc_kernel(const float* __restrict__ h1,
                                       const float* __restrict__ W_ih,
                                       const float* __restrict__ W_hh,
                                       const float* __restrict__ b_ih,
                                       const float* __restrict__ b_hh,
                                       const float* __restrict__ W_fc,
                                       const float* __restrict__ b_fc,
                                       float* __restrict__ out) {
    __shared__ float wip[32 * 256];    // [32][256] k-paired (K=64)
    __shared__ float whp[16 * 256];    // [16][256] k-paired (K=32)
    __shared__ float xbuf[16 * H1D];
    __shared__ float hbuf[16 * H2D];

    const int tid = threadIdx.x, nthr = blockDim.x;   // 64
    for (int idx = tid; idx < 32 * 256; idx += nthr) {
        int kp = idx >> 8, rem = idx & 255;
        int n = rem >> 1, k = kp * 2 + (rem & 1);
        wip[idx] = W_ih[n * H1D + k];
    }
    for (int idx = tid; idx < 16 * 256; idx += nthr) {
        int kp = idx >> 8, rem = idx & 255;
        int n = rem >> 1, k = kp * 2 + (rem & 1);
        whp[idx] = W_hh[n * H2D + k];
    }
    for (int idx = tid; idx < 16 * H2D; idx += nthr) hbuf[idx] = 0.0f;
    __syncthreads();

    const int wave = tid >> 5, lane = tid & 31;
    const int l15 = lane & 15, lhalf = lane >> 4;
    const int hid0 = wave * 16;
    const int b0 = blockIdx.x * 16;

    const int coli = 0 * H2D + hid0 + l15;
    const int colf = 1 * H2D + hid0 + l15;
    const int colg = 2 * H2D + hid0 + l15;
    const int colo = 3 * H2D + hid0 + l15;

    const float bias_i = b_ih[coli] + b_hh[coli];
    const float bias_f = b_ih[colf] + b_hh[colf];
    const float bias_g = b_ih[colg] + b_hh[colg];
    const float bias_o = b_ih[colo] + b_hh[colo];

    v8f cstate;
    #pragma unroll
    for (int r = 0; r < 8; ++r) cstate[r] = 0.0f;

    for (int t = 0; t < T_SEQ; ++t) {
        for (int i = tid; i < 16 * (H1D / 4); i += nthr) {
            int r = i >> 4, c4 = i & 15;
            ((v4f*)xbuf)[i] =
                ((const v4f*)h1)[((size_t)(b0 + r) * T_SEQ + t) * (H1D / 4) + c4];
        }
        __syncthreads();

        v8f ai, af, ag, ao;
        #pragma unroll
        for (int r = 0; r < 8; ++r) { ai[r] = bias_i; af[r] = bias_f; ag[r] = bias_g; ao[r] = bias_o; }

        for (int ks = 0; ks < H1D / 4; ++ks) {
            const int kp = ks * 2 + lhalf;
            v2f a = *(const v2f*)(xbuf + l15 * H1D + kp * 2);
            ai = wmma4(a, *(const v2f*)(wip + kp * 256 + 2 * coli), ai);
            af = wmma4(a, *(const v2f*)(wip + kp * 256 + 2 * colf), af);
            ag = wmma4(a, *(const v2f*)(wip + kp * 256 + 2 * colg), ag);
            ao = wmma4(a, *(const v2f*)(wip + kp * 256 + 2 * colo), ao);
        }
        for (int ks = 0; ks < H2D / 4; ++ks) {
            const int kp = ks * 2 + lhalf;
            v2f a = *(const v2f*)(hbuf + l15 * H2D + kp * 2);
            ai = wmma4(a, *(const v2f*)(whp + kp * 256 + 2 * coli), ai);
            af = wmma4(a, *(const v2f*)(whp + kp * 256 + 2 * colf), af);
            ag = wmma4(a, *(const v2f*)(whp + kp * 256 + 2 * colg), ag);
            ao = wmma4(a, *(const v2f*)(whp + kp * 256 + 2 * colo), ao);
        }
        __syncthreads();

        #pragma unroll
        for (int r = 0; r < 8; ++r) {
            float iv = sigmoid_(ai[r]);
            float fv = sigmoid_(af[r]);
            float gv = tanhf(ag[r]);
            float ov = sigmoid_(ao[r]);
            float cv = fv * cstate[r] + iv * gv;
            cstate[r] = cv;
            hbuf[(r + 8 * lhalf) * H2D + hid0 + l15] = ov * tanhf(cv);
        }
        __syncthreads();
    }

    const int m = tid >> 2, cls = tid & 3;
    float s = b_fc[cls];
    #pragma unroll
    for (int j = 0; j < H2D; ++j) s += hbuf[m * H2D + j] * W_fc[cls * H2D + j];
    out[(size_t)(b0 + m) * NCLS + cls] = s;
}

// ===========================================================================
extern "C" void kernel_launch(void* const* d_in, const int* in_sizes, int n_in,
                              void* d_out, int out_size, void* d_ws, size_t ws_size,
                              hipStream_t stream) {
    const float* x     = (const float*)d_in[0];
    const float* W_ih1 = (const float*)d_in[1];
    const float* W_hh1 = (const float*)d_in[2];
    const float* b_ih1 = (const float*)d_in[3];
    const float* b_hh1 = (const float*)d_in[4];
    const float* W_ih2 = (const float*)d_in[5];
    const float* W_hh2 = (const float*)d_in[6];
    const float* b_ih2 = (const float*)d_in[7];
    const float* b_hh2 = (const float*)d_in[8];
    const float* W_fc  = (const float*)d_in[9];
    const float* b_fc  = (const float*)d_in[10];
    float* out = (float*)d_out;

    const size_t n_xp1 = (size_t)MROWS * G1;   // 537 MB
    const size_t n_h1  = (size_t)MROWS * H1D;  // 134 MB
    const size_t n_xp2 = (size_t)MROWS * G2;   // 268 MB
    const size_t need_split = (n_xp1 + n_h1 + n_xp2) * sizeof(float);

    if (ws_size >= need_split) {
        // --- split path: parallel projections + short-critical-path scans ---
        float* xp1 = (float*)d_ws;
        float* h1  = xp1 + n_xp1;
        float* xp2 = h1 + n_h1;

        const size_t lds_p1 = (size_t)(66 * 512 + 32 * IN_PAD) * sizeof(float);
        const size_t lds_p2 = (size_t)(32 * 256 + 32 * H1D) * sizeof(float);
        const size_t lds_r1 = (size_t)(32 * 512 + 16 * G1 + 16 * H1D) * sizeof(float);

        proj_kernel<IN_DIM, IN_PAD, G1, 4><<<2048, 256, lds_p1, stream>>>(
            x, W_ih1, b_ih1, b_hh1, xp1, MROWS / 32);
        rec1_kernel<<<BATCH / 16, 128, lds_r1, stream>>>(xp1, W_hh1, h1);
        proj_kernel<H1D, H1D, G2, 2><<<2048, 256, lds_p2, stream>>>(
            h1, W_ih2, b_ih2, b_hh2, xp2, MROWS / 32);
        rec2_fc_kernel<<<BATCH / 16, 64, 0, stream>>>(xp2, W_hh2, W_fc, b_fc, out);
    } else {
        // --- fused fallback: only h1 (134 MB) in workspace ---
        float* h1 = (float*)d_ws;
        const size_t lds_f1 =
            (size_t)(66 * 512 + 32 * 512 + 16 * IN_PAD + 16 * H1D) * sizeof(float);
        fused_layer1_kernel<<<BATCH / 16, 128, lds_f1, stream>>>(
            x, W_ih1, W_hh1, b_ih1, b_hh1, h1);
        fused_layer2_fc_kernel<<<BATCH / 16, 64, 0, stream>>>(
            h1, W_ih2, W_hh2, b_ih2, b_hh2, W_fc, b_fc, out);
    }
}